// CAN_51539607710
// MI455X (gfx1250) — compile-verified
//
#include <hip/hip_runtime.h>
#include <hip/hip_bf16.h>

// CAN per-sample tiny-MLP: x(50x16) -> relu(x@W0+b0) -> relu(.@W1+b1)
// One wave32 per sample, fp32 WMMA 16x16x4 on gfx1250.

typedef float v2f __attribute__((ext_vector_type(2)));
typedef float v8f __attribute__((ext_vector_type(8)));

#define DIM 16
#define NSEQ 50
#define NTILE 4                    // 64 padded rows / 16
#define LAYERS 2
#define ITEM_STRIDE (DIM * (DIM + 1) * LAYERS)   // 544 floats per sample
#define WPB 8                      // waves (samples) per block

__global__ __launch_bounds__(WPB * 32)
void can_wmma_kernel(const float* __restrict__ user_emb,
                     const float* __restrict__ item_emb,
                     float* __restrict__ out,
                     int B)
{
    // per-wave 64x16 fp32 slab for the inter-layer layout bounce (D-layout -> A-layout)
    __shared__ float lds_x[WPB][64 * DIM];

    const int tid    = threadIdx.x;
    const int wave   = tid >> 5;
    const int lane   = tid & 31;
    const int lane16 = lane & 15;
    const int hi     = lane >> 4;          // 0 for lanes 0-15, 1 for lanes 16-31

    const int b = blockIdx.x * WPB + wave; // sample index (wave-uniform)
    if (b >= B) return;

    const float* __restrict__ xg = user_emb + (size_t)b * (NSEQ * DIM);
    const float* __restrict__ ip = item_emb + (size_t)b * ITEM_STRIDE;
    float* __restrict__ og       = out      + (size_t)b * (NSEQ * DIM);
    float* lx = lds_x[wave];

#pragma unroll
    for (int layer = 0; layer < LAYERS; ++layer) {
        const float* __restrict__ Wp = ip + layer * (DIM * DIM + DIM);

        // ---- B-matrix (W, 16x16) in WMMA B-layout: 4 K-chunks x 2 VGPRs ----
        // chunk kc, vgpr j: lanes 0-15 hold (K = 4kc+j,   N = lane),
        //                   lanes16-31 hold (K = 4kc+j+2, N = lane-16)
        v2f Bk[4];
#pragma unroll
        for (int kc = 0; kc < 4; ++kc) {
            const int krow = 4 * kc + 2 * hi;
            Bk[kc].x = Wp[(krow + 0) * DIM + lane16];
            Bk[kc].y = Wp[(krow + 1) * DIM + lane16];
        }
        const float bias = Wp[DIM * DIM + lane16];   // per-N bias

        // ---- 4 row tiles of 16 ----
#pragma unroll
        for (int rt = 0; rt < NTILE; ++rt) {
            const int r  = rt * 16 + lane16;
            const int rc = (r < NSEQ) ? r : (NSEQ - 1);   // clamp pad rows (no OOB)

            v8f acc = {0.f, 0.f, 0.f, 0.f, 0.f, 0.f, 0.f, 0.f};

#pragma unroll
            for (int kc = 0; kc < 4; ++kc) {
                // A-layout: lane holds 2 consecutive elements of its row:
                // (M = lane16, K = 4kc + 2*hi) and (K+1)  -> one 8-byte load
                const int col = 4 * kc + 2 * hi;
                v2f a;
                if (layer == 0) {
                    a = *(const v2f*)(xg + rc * DIM + col);
                } else {
                    a = *(const v2f*)(lx + rc * DIM + col);
                }
                acc = __builtin_amdgcn_wmma_f32_16x16x4_f32(
                        /*neg_a=*/false, a,
                        /*neg_b=*/false, Bk[kc],
                        /*c_mod=*/(short)0, acc,
                        /*reuse_a=*/false, /*reuse_b=*/false);
            }

            // ---- bias + relu on D-layout, then scatter ----
            // D-layout: VGPR v, lane -> element (M = rt*16 + v + 8*hi, N = lane16)
#pragma unroll
            for (int v = 0; v < 8; ++v) {
                float val = acc[v] + bias;
                val = (val > 0.f) ? val : 0.f;
                const int row = rt * 16 + v + 8 * hi;
                if (layer == 0) {
                    lx[row * DIM + lane16] = val;     // all 64 rows (pads unread)
                } else if (row < NSEQ) {
                    og[row * DIM + lane16] = val;     // only valid rows
                }
            }
        }
    }
}

extern "C" void kernel_launch(void* const* d_in, const int* in_sizes, int n_in,
                              void* d_out, int out_size, void* d_ws, size_t ws_size,
                              hipStream_t stream) {
    const float* user_emb = (const float*)d_in[0];
    const float* item_emb = (const float*)d_in[1];
    float* out = (float*)d_out;

    const int B = in_sizes[0] / (NSEQ * DIM);   // 16384
    const int grid = (B + WPB - 1) / WPB;

    can_wmma_kernel<<<grid, WPB * 32, 0, stream>>>(user_emb, item_emb, out, B);
}